// FrameLoss_87729001988727
// MI455X (gfx1250) — compile-verified
//
#include <hip/hip_runtime.h>

// MI455X / gfx1250, wave32.
// sum((x - target)^2) = sum(x^2) - 2*sum(t*x) + sum(t^2); target has <=4 nnz/row.
// Kernel A: HBM-bound sum of squares (134 MB stream; ~5.8us floor at 23.3 TB/s).
//           4x-unrolled b128 loads (4 independent loads in flight per wave ->
//           ~16 MB chip-wide MLP, enough to cover DRAM latency at 23.3 TB/s).
//           Wave-level horizontal reduction via V_WMMA_F32_16X16X4_F32 with
//           B = ones: one instruction collapses all 64 per-wave accumulators.
// Kernel B: per-row sparse correction (<=4 gathers/row, 32K total loads).
// Kernel C: deterministic final tree reduction (no float atomics).

typedef __attribute__((ext_vector_type(2))) float v2f;
typedef __attribute__((ext_vector_type(8))) float v8f;

#define BLOCK 256
#define WAVES_PER_BLOCK (BLOCK / 32)
#define GRID_A 1024

__global__ __launch_bounds__(BLOCK)
void sumsq_kernel(const float* __restrict__ x, size_t total,
                  float* __restrict__ partials) {
    __shared__ float lds[WAVES_PER_BLOCK];
    const size_t num4   = total >> 2;
    const size_t tid    = (size_t)blockIdx.x * BLOCK + threadIdx.x;
    const size_t stride = (size_t)gridDim.x * BLOCK;
    const float4* x4 = (const float4*)x;

    // 4 independent accumulator pairs -> short FMA chains, loads never stalled.
    float a0x = 0.f, a0y = 0.f, a1x = 0.f, a1y = 0.f;
    float a2x = 0.f, a2y = 0.f, a3x = 0.f, a3y = 0.f;

    size_t i = tid;
    for (; i + 3 * stride < num4; i += 4 * stride) {
        float4 v0 = x4[i];                 // 4 independent global_load_b128
        float4 v1 = x4[i + stride];
        float4 v2 = x4[i + 2 * stride];
        float4 v3 = x4[i + 3 * stride];
        a0x = fmaf(v0.x, v0.x, a0x); a0y = fmaf(v0.y, v0.y, a0y);
        a0x = fmaf(v0.z, v0.z, a0x); a0y = fmaf(v0.w, v0.w, a0y);
        a1x = fmaf(v1.x, v1.x, a1x); a1y = fmaf(v1.y, v1.y, a1y);
        a1x = fmaf(v1.z, v1.z, a1x); a1y = fmaf(v1.w, v1.w, a1y);
        a2x = fmaf(v2.x, v2.x, a2x); a2y = fmaf(v2.y, v2.y, a2y);
        a2x = fmaf(v2.z, v2.z, a2x); a2y = fmaf(v2.w, v2.w, a2y);
        a3x = fmaf(v3.x, v3.x, a3x); a3y = fmaf(v3.y, v3.y, a3y);
        a3x = fmaf(v3.z, v3.z, a3x); a3y = fmaf(v3.w, v3.w, a3y);
    }
    for (; i < num4; i += stride) {        // ragged remainder (empty for 8192x4096)
        float4 v = x4[i];
        a0x = fmaf(v.x, v.x, a0x); a0y = fmaf(v.y, v.y, a0y);
        a0x = fmaf(v.z, v.z, a0x); a0y = fmaf(v.w, v.w, a0y);
    }
    // scalar tail if total % 4 != 0; one lane, reconverges before WMMA
    if (blockIdx.x == 0 && threadIdx.x == 0) {
        for (size_t j = num4 << 2; j < total; ++j) {
            float v = x[j];
            a0x = fmaf(v, v, a0x);
        }
    }

    // --- WMMA horizontal reduction of the 64 per-wave accumulators ---
    // A (16x4 f32) = merged pairs across 32 lanes; B (4x16) = ones; C = 0.
    // D[m][n] = sum_k A[m][k] (independent of n). EXEC is all-ones here.
    v2f A; A.x = (a0x + a1x) + (a2x + a3x);
           A.y = (a0y + a1y) + (a2y + a3y);
    v2f B; B.x = 1.0f; B.y = 1.0f;
    v8f C = {};
    C = __builtin_amdgcn_wmma_f32_16x16x4_f32(
        /*neg_a=*/false, A, /*neg_b=*/false, B,
        /*c_mod=*/(short)0, C, /*reuse_a=*/false, /*reuse_b=*/false);

    // Per lane: sum of its 8 D entries = 8 rows of one column.
    float s = C[0] + C[1] + C[2] + C[3] + C[4] + C[5] + C[6] + C[7];
    // lanes 0-15 hold rows 0-7, lanes 16-31 rows 8-15 (columns identical):
    s += __shfl_xor(s, 16, 32);            // full wave total in every lane

    const int wave = threadIdx.x >> 5;
    const int lane = threadIdx.x & 31;
    if (lane == 0) lds[wave] = s;
    __syncthreads();
    if (threadIdx.x == 0) {
        float t = 0.0f;
        for (int w = 0; w < WAVES_PER_BLOCK; ++w) t += lds[w];
        partials[blockIdx.x] = t;
    }
}

__global__ __launch_bounds__(BLOCK)
void corr_kernel(const float* __restrict__ x, const float* __restrict__ label,
                 int N, int T, float* __restrict__ partials) {
    __shared__ float lds[BLOCK];
    const int i = blockIdx.x * BLOCK + threadIdx.x;
    float c = 0.0f;
    if (i < N) {
        float pos = label[i] - 1.0f;          // in [0, T-1)
        int fl = (int)floorf(pos);
        int cl = (int)ceilf(pos);
        const float* row = x + (size_t)i * (size_t)T;
        float sumT2 = (fl == cl) ? 1.0f : 2.0f;                 // weight-1 hits
        float sumTX = row[fl] + ((fl == cl) ? 0.0f : row[cl]);
        if (fl >= 1)     { sumT2 += 0.25f; sumTX += 0.5f * row[fl - 1]; }
        if (cl < T - 1)  { sumT2 += 0.25f; sumTX += 0.5f * row[cl + 1]; }
        c = sumT2 - 2.0f * sumTX;             // sum(t^2) - 2*sum(t*x) for this row
    }
    lds[threadIdx.x] = c;
    __syncthreads();
    for (int off = BLOCK / 2; off > 0; off >>= 1) {
        if (threadIdx.x < off) lds[threadIdx.x] += lds[threadIdx.x + off];
        __syncthreads();
    }
    if (threadIdx.x == 0) partials[blockIdx.x] = lds[0];
}

__global__ __launch_bounds__(BLOCK)
void final_kernel(const float* __restrict__ partials, int n,
                  float* __restrict__ out) {
    __shared__ float lds[BLOCK];
    float s = 0.0f;
    for (int i = threadIdx.x; i < n; i += BLOCK) s += partials[i];
    lds[threadIdx.x] = s;
    __syncthreads();
    for (int off = BLOCK / 2; off > 0; off >>= 1) {
        if (threadIdx.x < off) lds[threadIdx.x] += lds[threadIdx.x + off];
        __syncthreads();
    }
    if (threadIdx.x == 0) out[0] = lds[0];
}

extern "C" void kernel_launch(void* const* d_in, const int* in_sizes, int n_in,
                              void* d_out, int out_size, void* d_ws, size_t ws_size,
                              hipStream_t stream) {
    const float* x     = (const float*)d_in[0];   // [N, T] f32
    const float* label = (const float*)d_in[1];   // [N]    f32
    float* out = (float*)d_out;                   // scalar f32
    float* ws  = (float*)d_ws;

    const size_t total = (size_t)in_sizes[0];
    const int N = in_sizes[1];
    const int T = (int)(total / (size_t)N);
    const int gridB = (N + BLOCK - 1) / BLOCK;

    // ws layout: [0, GRID_A) bulk partials, [GRID_A, GRID_A+gridB) correction partials
    sumsq_kernel<<<GRID_A, BLOCK, 0, stream>>>(x, total, ws);
    corr_kernel<<<gridB, BLOCK, 0, stream>>>(x, label, N, T, ws + GRID_A);
    final_kernel<<<1, BLOCK, 0, stream>>>(ws, GRID_A + gridB, out);
}